// Enframe_21380347199933
// MI455X (gfx1250) — compile-verified
//
#include <hip/hip_runtime.h>

// ---- Problem constants (from reference) ----
#define FL     2048
#define HOP    512
#define S_LEN  480000
#define B_N    8
#define C_N    2
#define T_N    (((S_LEN) - (FL)) / (HOP) + 1)   // 934

// ---- Tiling ----
#define TK         64          // k values per tile
#define TT         64          // t values per tile
#define LDS_STRIDE 65          // 64 + 1 pad: stride ≡ 1 (mod 64) -> conflict-free column reads
#define NTHREADS   256         // 8 wave32

typedef __attribute__((address_space(1))) int g_int;   // global (device) int
typedef __attribute__((address_space(3))) int l_int;   // LDS int
typedef __attribute__((address_space(3))) void lds_void;

// LDS byte offset (relative to workgroup LDS base) of a generic shared pointer.
static __device__ __forceinline__ unsigned lds_offset_of(const void* p) {
    return (unsigned)(unsigned long long)(lds_void*)p;
}

// Async copy: 4 bytes per lane, global -> LDS. CDNA5 GLOBAL_LOAD_ASYNC_TO_LDS_B32.
static __device__ __forceinline__ void async_g2l_b32(const float* gp, float* lp) {
#if __has_builtin(__builtin_amdgcn_global_load_async_to_lds_b32)
    __builtin_amdgcn_global_load_async_to_lds_b32(
        (g_int*)gp, (l_int*)lp, 0, 0);
#else
    unsigned long long ga = (unsigned long long)gp;
    unsigned loff = lds_offset_of(lp);
    asm volatile("global_load_async_to_lds_b32 %0, %1, off"
                 :: "v"(loff), "v"(ga)
                 : "memory");
#endif
}

static __device__ __forceinline__ void wait_async0() {
#if __has_builtin(__builtin_amdgcn_s_wait_asynccnt)
    __builtin_amdgcn_s_wait_asynccnt(0);
#else
    asm volatile("s_wait_asynccnt 0" ::: "memory");
#endif
}

__global__ __launch_bounds__(NTHREADS)
void enframe_tile_kernel(const float* __restrict__ in, float* __restrict__ out) {
    __shared__ float tile[TT * LDS_STRIDE];   // tile[tt * 65 + kk]

    const int t0  = blockIdx.x * TT;          // 15 tiles over t (last partial)
    const int k0  = blockIdx.y * TK;          // 32 tiles over k
    const int p   = blockIdx.z;               // plane = b*C + c   (16 planes)
    const int tid = threadIdx.x;

    const float* in_plane = in + (size_t)p * S_LEN;

    // ---- Phase 1: global -> LDS, async, fully coalesced reads ----
    // 64 rows (tt) x 64 floats (kk) = 4096 elems; 16 async b32 per thread.
    // Lanes 0..31 of a wave cover 32 consecutive kk at one tt -> 128B contiguous.
#pragma unroll
    for (int m = 0; m < (TT * TK) / NTHREADS; ++m) {
        const int flat = m * NTHREADS + tid;
        const int kk   = flat & (TK - 1);
        const int tt   = flat >> 6;
        int t = t0 + tt;
        if (t > T_N - 1) t = T_N - 1;         // clamp: stays in-bounds, dup rows unused
        const float* gp = in_plane + (size_t)t * HOP + (size_t)(k0 + kk);
        async_g2l_b32(gp, &tile[tt * LDS_STRIDE + kk]);
    }

    wait_async0();        // this wave's async transfers landed in LDS
    __syncthreads();      // make all waves' tiles visible

    // ---- Phase 2: LDS -> global, float2 per thread, fully coalesced writes ----
    // Whole wave writes one output row's 64 t-values (256B contiguous).
    // LDS column reads: bank = (2*lane + j + krow) mod 64 -> 32 distinct banks.
    float* out_plane = out + (size_t)p * (size_t)FL * T_N;
#pragma unroll
    for (int m = 0; m < (TT * TK) / (2 * NTHREADS); ++m) {
        const int flat = m * NTHREADS + tid;
        const int tv   = flat & 31;           // lane id within wave
        const int kr   = flat >> 5;           // same for all lanes of a wave
        const int t    = t0 + 2 * tv;
        if (t < T_N) {                        // T_N and t0 even: pair never straddles
            float2 v;
            v.x = tile[(2 * tv + 0) * LDS_STRIDE + kr];
            v.y = tile[(2 * tv + 1) * LDS_STRIDE + kr];
            // row*T_N + t is even -> 8B-aligned b64 store
            *reinterpret_cast<float2*>(out_plane + (size_t)(k0 + kr) * T_N + t) = v;
        }
    }
}

extern "C" void kernel_launch(void* const* d_in, const int* in_sizes, int n_in,
                              void* d_out, int out_size, void* d_ws, size_t ws_size,
                              hipStream_t stream) {
    (void)in_sizes; (void)n_in; (void)out_size; (void)d_ws; (void)ws_size;
    const float* x = (const float*)d_in[0];
    float* out    = (float*)d_out;
    dim3 grid((T_N + TT - 1) / TT,   // 15
              FL / TK,               // 32
              B_N * C_N);            // 16
    enframe_tile_kernel<<<grid, NTHREADS, 0, stream>>>(x, out);
}